// AttentionKVCache_32933809226222
// MI455X (gfx1250) — compile-verified
//
#include <hip/hip_runtime.h>
#include <hip/hip_bf16.h>
#include <math.h>

#define EMB_DIM 4096
#define OUT_DIM 4096
#define T_PREV  8192

#define KRANGE  2048    // K-range owned by one wave (per blockIdx.y)
#define TK      256     // K-width of one TDM sub-tile
#define TKP     260     // padded LDS row stride (TDM pads 4 dwords / 256 dwords)
#define NSUB    (KRANGE / TK)   // 8 sub-tiles, double-buffered

typedef __attribute__((ext_vector_type(2))) float    v2f;
typedef __attribute__((ext_vector_type(4))) float    v4f;
typedef __attribute__((ext_vector_type(8))) float    v8f;
typedef __attribute__((ext_vector_type(4))) unsigned su4;
typedef __attribute__((ext_vector_type(8))) unsigned su8;

// ---------------------------------------------------------------------------
// WMMA f32 GEMV partial with TDM tile staging. One wave per block.
// Block (bx,by): rows [bx*16, bx*16+16) of row-major A[nRows x rowStride],
// partial dots against x over K-range [by*KRANGE, (by+1)*KRANGE).
//   partial[by*nRows + row] = sum_k A[row][k] * x[k]
// A tiles (16 x TK f32) are DMA'd global->LDS by TENSOR_LOAD_TO_LDS with
// 4-dword padding every 256 dwords -> LDS row stride 260 floats
// (260 mod 64 == 4: the 32-lane b64 fragment read covers all 64 banks,
// conflict-free). Two tiles are kept in flight on TENSORcnt.
// WMMA layouts per ISA 7.12.2 (32-bit, 16x16x4):
//   A: lane m(0-15): A[m][k],A[m][k+1]; lane 16+m: A[m][k+2],A[m][k+3]
//   B: replicated x across all N columns -> every D column is the result
//   C/D: vgpr r: lanes 0-15 -> M=r, lanes 16-31 -> M=8+r
// ---------------------------------------------------------------------------
__global__ __launch_bounds__(32) void gemv16_wmma_tdm(
    const float* __restrict__ A, const float* __restrict__ x,
    float* __restrict__ partial, int nRows, int rowStride)
{
  __shared__ float tiles[2][16 * TKP];   // 2 x 16,640 B (incl. trailing pad)
  __shared__ float xs[KRANGE];           // 8 KB x stage

  const int lane = threadIdx.x;
  const int h    = lane >> 4;            // half-wave selects K pair
  const int m    = lane & 15;            // row within 16-row tile
  const int row0 = blockIdx.x * 16;
  const int k0   = blockIdx.y * KRANGE;

  const unsigned ldsTile0 = (unsigned)(size_t)&tiles[0][0];
  const unsigned ldsTile1 = (unsigned)(size_t)&tiles[1][0];
  const unsigned long long gbase =
      (unsigned long long)(size_t)(A + (size_t)row0 * rowStride + k0);

  // D# group 1 (ISA 8.4): data_size=4B, pad_enable, pad_interval=7 (256 dw),
  // pad_amount=3 (4 dw), tensor 256x16, tile 256x16, dim0 stride = rowStride.
  su8 g1;
  g1[0] = (2u << 16) | (1u << 20) | (7u << 22) | (3u << 25);
  g1[1] = ((unsigned)TK) << 16;          // tensor_dim0[15:0]
  g1[2] = (16u << 16);                   // tensor_dim0 hi=0 | tensor_dim1=16
  g1[3] = ((unsigned)TK) << 16;          // tensor_dim1 hi=0 | tile_dim0=TK
  g1[4] = 16u;                           // tile_dim1=16, tile_dim2=0 (2D)
  g1[5] = (unsigned)rowStride;           // tensor_dim0_stride lo32 (elements)
  g1[6] = 0u;
  g1[7] = 0u;

  auto issue = [&](int s, unsigned ldsDst) {
    unsigned long long ga = gbase + (unsigned long long)s * (TK * 4);
    su4 g0;
    g0[0] = 1u;                                   // count=1 (2D descriptor)
    g0[1] = ldsDst;                               // lds_addr (bytes)
    g0[2] = (unsigned)ga;                         // global_addr[31:0]
    g0[3] = (unsigned)(ga >> 32) | 0x80000000u;   // global_addr[56:32] | type=2
    asm volatile("tensor_load_to_lds %0, %1" :: "s"(g0), "s"(g1) : "memory");
  };

  // Kick off the first two tile DMAs, then stage x while they fly.
  issue(0, ldsTile0);
  issue(1, ldsTile1);
  {
    const v4f* xg = (const v4f*)(x + k0);
    v4f* xl = (v4f*)xs;
#pragma unroll
    for (int i = 0; i < KRANGE / 4 / 32; ++i)
      xl[lane + 32 * i] = xg[lane + 32 * i];
  }
  __syncthreads();

  const int aoff = m * TKP + 2 * h;
  v8f acc = {0.f, 0.f, 0.f, 0.f, 0.f, 0.f, 0.f, 0.f};

  auto process = [&](int s, int buf) {
    const int xoff = s * TK + 2 * h;
#pragma unroll 4
    for (int k = 0; k < TK; k += 4) {
      v2f av = *(const v2f*)&tiles[buf][aoff + k];
      v2f bv = *(const v2f*)&xs[xoff + k];
      acc = __builtin_amdgcn_wmma_f32_16x16x4_f32(
          false, av, false, bv, (short)0, acc, false, false);
    }
  };

#pragma unroll
  for (int s = 0; s < NSUB - 1; ++s) {
    __builtin_amdgcn_s_wait_tensorcnt(1);   // tile s landed
    process(s, s & 1);
    if (s + 2 < NSUB) {
      asm volatile("s_wait_dscnt 0x0" ::: "memory");  // tile consumed
      issue(s + 2, (s & 1) ? ldsTile1 : ldsTile0);
    }
  }
  __builtin_amdgcn_s_wait_tensorcnt(0);
  process(NSUB - 1, (NSUB - 1) & 1);

  if (m == 0) {  // lanes 0 (rows 0-7) and 16 (rows 8-15)
    float* out = partial + (size_t)blockIdx.y * nRows + row0 + 8 * h;
#pragma unroll
    for (int r = 0; r < 8; ++r) out[r] = acc[r];
  }
}

// Sum the 2 K-chunk partials for each of q,k,v (fixed order -> deterministic).
__global__ __launch_bounds__(256) void qkv_combine_kernel(
    const float* __restrict__ part, float* __restrict__ qkv)
{
  int i = blockIdx.x * 256 + threadIdx.x;     // 0..12287
  int mtx = i >> 12, r = i & 4095;
  const float* p = part + (size_t)mtx * (2 * OUT_DIM) + r;
  float s = 0.f;
#pragma unroll
  for (int c = 0; c < 2; ++c) s += p[(size_t)c * OUT_DIM];
  qkv[i] = s;
}

// scores partial: part[dchunk][t] = sum_{d in chunk} q[d] * K[d][t]
// lane = t -> 128B coalesced per wave per step; q chunk broadcast from LDS.
#define DCHUNK 512
__global__ __launch_bounds__(256) void scores_partial_kernel(
    const float* __restrict__ Kc, const float* __restrict__ q,
    float* __restrict__ part)
{
  __shared__ float qs[DCHUNK];
  const int t  = blockIdx.x * 256 + threadIdx.x;
  const int d0 = blockIdx.y * DCHUNK;
  for (int i = threadIdx.x; i < DCHUNK; i += 256) qs[i] = q[d0 + i];
  __syncthreads();

  const float* kp = Kc + (size_t)d0 * T_PREV + t;
  float acc = 0.f;
#pragma unroll 8
  for (int d = 0; d < DCHUNK; ++d) {
    acc = fmaf(qs[d], *kp, acc);
    kp += T_PREV;
  }
  part[(size_t)blockIdx.y * T_PREV + t] = acc;
}

// a[t] = sigmoid(sum_c part[c][t] / 64); block 32 does the appended column
// a[T_PREV] = sigmoid(dot(q,k)/64) via a fixed-order LDS reduction.
__global__ __launch_bounds__(256) void scores_combine_kernel(
    const float* __restrict__ part, const float* __restrict__ q,
    const float* __restrict__ k, float* __restrict__ a)
{
  const float scale = 0.015625f;  // 1/sqrt(4096)
  if (blockIdx.x < 32) {
    int t = blockIdx.x * 256 + threadIdx.x;
    float s = 0.f;
#pragma unroll
    for (int c = 0; c < 8; ++c) s += part[(size_t)c * T_PREV + t];
    s *= scale;
    a[t] = 1.f / (1.f + __expf(-s));
  } else {
    __shared__ float red[256];
    float s = 0.f;
    for (int i = threadIdx.x; i < OUT_DIM; i += 256)
      s = fmaf(q[i], k[i], s);
    red[threadIdx.x] = s;
    __syncthreads();
    for (int w = 128; w > 0; w >>= 1) {
      if (threadIdx.x < w) red[threadIdx.x] += red[threadIdx.x + w];
      __syncthreads();
    }
    if (threadIdx.x == 0) {
      float v = red[0] * scale;
      a[T_PREV] = 1.f / (1.f + __expf(-v));
    }
  }
}

// z[d] = sum_c zpart[c][d] + v[d] * a[T_PREV]
__global__ __launch_bounds__(256) void out_combine_kernel(
    const float* __restrict__ part, const float* __restrict__ v,
    const float* __restrict__ a, float* __restrict__ z)
{
  int d = blockIdx.x * 256 + threadIdx.x;   // 0..4095
  float s = 0.f;
#pragma unroll
  for (int c = 0; c < 4; ++c) s += part[(size_t)c * OUT_DIM + d];
  z[d] = fmaf(v[d], a[T_PREV], s);
}

extern "C" void kernel_launch(void* const* d_in, const int* in_sizes, int n_in,
                              void* d_out, int out_size, void* d_ws, size_t ws_size,
                              hipStream_t stream) {
  const float* x  = (const float*)d_in[0];
  const float* Wq = (const float*)d_in[1];
  const float* Wk = (const float*)d_in[2];
  const float* Wv = (const float*)d_in[3];
  const float* Kc = (const float*)d_in[4];
  const float* Vc = (const float*)d_in[5];
  float* z  = (float*)d_out;
  float* ws = (float*)d_ws;

  // workspace layout (floats)
  float* qkvPart = ws;               // 3 * 2 * 4096 = 24576
  float* qkv     = ws + 24576;       // 12288 (q | k | v)
  float* scPart  = ws + 36864;       // 8 * 8192 = 65536
  float* a       = ws + 102400;      // 8193 (16B-aligned base)
  float* zPart   = ws + 110596;      // 4 * 4096 = 16384

  const dim3 b32(32), b256(256);

  // Phase 1: q,k,v partials (grid 256 x 2 per matrix), then combine.
  gemv16_wmma_tdm<<<dim3(OUT_DIM / 16, EMB_DIM / KRANGE), b32, 0, stream>>>(
      Wq, x, qkvPart + 0 * 2 * OUT_DIM, OUT_DIM, EMB_DIM);
  gemv16_wmma_tdm<<<dim3(OUT_DIM / 16, EMB_DIM / KRANGE), b32, 0, stream>>>(
      Wk, x, qkvPart + 1 * 2 * OUT_DIM, OUT_DIM, EMB_DIM);
  gemv16_wmma_tdm<<<dim3(OUT_DIM / 16, EMB_DIM / KRANGE), b32, 0, stream>>>(
      Wv, x, qkvPart + 2 * 2 * OUT_DIM, OUT_DIM, EMB_DIM);
  qkv_combine_kernel<<<dim3(3 * OUT_DIM / 256), b256, 0, stream>>>(qkvPart, qkv);

  // Phase 2: sigmoid attention scores over K_cache + appended k.
  scores_partial_kernel<<<dim3(T_PREV / 256, EMB_DIM / DCHUNK), b256, 0, stream>>>(
      Kc, qkv /*q*/, scPart);
  scores_combine_kernel<<<dim3(33), b256, 0, stream>>>(
      scPart, qkv /*q*/, qkv + OUT_DIM /*k*/, a);

  // Phase 3: z = V_cache @ a[0:8192] (TDM + WMMA GEMV) + v * a[8192].
  gemv16_wmma_tdm<<<dim3(OUT_DIM / 16, T_PREV / KRANGE), b32, 0, stream>>>(
      Vc, a, zPart, OUT_DIM, T_PREV);
  out_combine_kernel<<<dim3(OUT_DIM / 256), b256, 0, stream>>>(
      zPart, qkv + 2 * OUT_DIM /*v*/, a, z);
}